// StochasticTwoLayerRGCN_78262894068345
// MI455X (gfx1250) — compile-verified
//
#include <hip/hip_runtime.h>

#define ND 50000
#define NP 50000
#define NE 600000
#define DIN 128
#define DH 128
#define DOUT2 64

typedef __attribute__((ext_vector_type(2))) float v2f;
typedef __attribute__((ext_vector_type(8))) float v8f;

// ---------------- utility: zero fill ----------------
__global__ void zero_kernel(float* __restrict__ p, size_t n) {
    size_t i = (size_t)blockIdx.x * blockDim.x + threadIdx.x;
    size_t stride = (size_t)gridDim.x * blockDim.x;
    for (; i < n; i += stride) p[i] = 0.0f;
}

// ---------------- in-degree (float, for normalization) ----------------
__global__ void degree_kernel(const int* __restrict__ dst, float* __restrict__ deg, int nE) {
    int i = blockIdx.x * blockDim.x + threadIdx.x;
    if (i < nE) unsafeAtomicAdd(&deg[dst[i]], 1.0f);
}

// ---------------- H[N,Do] = X[N,K] @ W[K,Do] via V_WMMA_F32_16X16X4_F32 ----------------
// One wave computes a 16-row x Do-column strip: the A fragment is loaded once per
// k-step and reused across Do/16 WMMAs (8x less X traffic than tile-per-wave).
// A 16x4 layout: lane<16 -> row=lane%16, K = k0+0 (v0), k0+1 (v1); lane>=16 -> K=k0+2,k0+3.
// B 4x16 layout: v0: K=k0+2*hi, v1: K=k0+2*hi+1, col = lane%16.
// C/D 16x16:     vgpr v: row = v + 8*(lane>>4), col = lane%16.
template <int K, int Do>
__global__ void gemm_wmma_f32(const float* __restrict__ X, const float* __restrict__ W,
                              float* __restrict__ H, int N) {
    constexpr int TN = Do / 16;
    int rowTile = blockIdx.x * (blockDim.x >> 5) + (threadIdx.x >> 5);
    if (rowTile >= (N >> 4)) return;
    int lane = threadIdx.x & 31;
    int lo = lane & 15;
    int hi = lane >> 4;

    const float* xr = X + (size_t)(rowTile * 16 + lo) * K;   // A row for this lane

    v8f acc[TN];
#pragma unroll
    for (int t = 0; t < TN; ++t) acc[t] = (v8f){};

    for (int k0 = 0; k0 < K; k0 += 4) {
        int ka = k0 + 2 * hi;
        v2f a;
        a.x = xr[ka];
        a.y = xr[ka + 1];
        const float* w0 = W + (size_t)ka * Do + lo;
        const float* w1 = W + (size_t)(ka + 1) * Do + lo;
#pragma unroll
        for (int t = 0; t < TN; ++t) {
            v2f b;
            b.x = w0[t * 16];
            b.y = w1[t * 16];
            acc[t] = __builtin_amdgcn_wmma_f32_16x16x4_f32(
                        /*neg_a=*/false, a, /*neg_b=*/false, b,
                        /*c_mod=*/(short)0, acc[t],
                        /*reuse_a=*/false, /*reuse_b=*/false);
        }
    }

    float* hr = H + (size_t)(rowTile * 16) * Do + lo;
#pragma unroll
    for (int t = 0; t < TN; ++t)
#pragma unroll
        for (int v = 0; v < 8; ++v)
            hr[(size_t)(v + 8 * hi) * Do + t * 16] = acc[t][v];
}

// ---------------- warp-per-edge scatter add: agg[dst] += h[src] ----------------
// Prefetch the next edge's source row (global_prefetch_b8) while issuing the
// current edge's native f32 atomics; this phase is the HBM-roofline bottleneck.
template <int D>
__global__ void scatter_add(const float* __restrict__ h, const int* __restrict__ src,
                            const int* __restrict__ dst, float* __restrict__ agg, int nE) {
    int warp = (blockIdx.x * blockDim.x + threadIdx.x) >> 5;
    int lane = threadIdx.x & 31;
    int nwarps = (gridDim.x * blockDim.x) >> 5;
    for (int e = warp; e < nE; e += nwarps) {
        int s = src[e];
        int d = dst[e];
        int e2 = e + nwarps;
        if (e2 < nE) {
            // lanes 16B apart cover the whole next source row
            const float* nxt = h + (size_t)src[e2] * D;
            __builtin_prefetch(nxt + lane * 4, 0, 3);
        }
        const float* hs = h + (size_t)s * D;
        float* ad = agg + (size_t)d * D;
#pragma unroll
        for (int i = 0; i < D / 32; ++i)
            unsafeAtomicAdd(&ad[lane + 32 * i], hs[lane + 32 * i]);
    }
}

// ---------------- out = agg / max(deg,1) + b  (single relation) ----------------
template <int D>
__global__ void finalize1(const float* __restrict__ agg, const float* __restrict__ deg,
                          const float* __restrict__ b, float* __restrict__ out, int n) {
    size_t i = (size_t)blockIdx.x * blockDim.x + threadIdx.x;
    size_t total = (size_t)n * D;
    if (i >= total) return;
    int row = (int)(i / D);
    int col = (int)(i - (size_t)row * D);
    out[i] = agg[i] / fmaxf(deg[row], 1.0f) + b[col];
}

// ---------------- out = aggA/max(degA,1)+bA + aggB/max(degB,1)+bB ----------------
template <int D>
__global__ void finalize2(const float* __restrict__ aggA, const float* __restrict__ degA,
                          const float* __restrict__ bA,
                          const float* __restrict__ aggB, const float* __restrict__ degB,
                          const float* __restrict__ bB,
                          float* __restrict__ out, int n) {
    size_t i = (size_t)blockIdx.x * blockDim.x + threadIdx.x;
    size_t total = (size_t)n * D;
    if (i >= total) return;
    int row = (int)(i / D);
    int col = (int)(i - (size_t)row * D);
    out[i] = aggA[i] / fmaxf(degA[row], 1.0f) + bA[col]
           + aggB[i] / fmaxf(degB[row], 1.0f) + bB[col];
}

extern "C" void kernel_launch(void* const* d_in, const int* in_sizes, int n_in,
                              void* d_out, int out_size, void* d_ws, size_t ws_size,
                              hipStream_t stream) {
    const float* x_drug = (const float*)d_in[0];
    const float* x_prot = (const float*)d_in[1];
    const int* src_ddi = (const int*)d_in[2];
    const int* dst_ddi = (const int*)d_in[3];
    const int* src_dpi = (const int*)d_in[4];
    const int* dst_dpi = (const int*)d_in[5];
    const int* src_ppi = (const int*)d_in[6];
    const int* dst_ppi = (const int*)d_in[7];
    const float* W1_ddi = (const float*)d_in[8];
    const float* b1_ddi = (const float*)d_in[9];
    const float* W1_dpi = (const float*)d_in[10];
    const float* b1_dpi = (const float*)d_in[11];
    const float* W1_ppi = (const float*)d_in[12];
    const float* b1_ppi = (const float*)d_in[13];
    const float* W2_ddi = (const float*)d_in[14];
    const float* b2_ddi = (const float*)d_in[15];
    const float* W2_dpi = (const float*)d_in[16];
    const float* b2_dpi = (const float*)d_in[17];
    const float* W2_ppi = (const float*)d_in[18];
    const float* b2_ppi = (const float*)d_in[19];
    float* out = (float*)d_out;

    // ---- workspace layout (floats). Zeroed region first (agg + degree). ----
    float* w = (float*)d_ws;
    size_t o = 0;
    float* aggd   = w + o; o += (size_t)ND * DH;
    float* aggp1  = w + o; o += (size_t)NP * DH;
    float* aggp2  = w + o; o += (size_t)NP * DH;
    float* agg2d  = w + o; o += (size_t)ND * DOUT2;
    float* agg2p1 = w + o; o += (size_t)NP * DOUT2;
    float* agg2p2 = w + o; o += (size_t)NP * DOUT2;
    float* deg_ddi = w + o; o += ND;
    float* deg_dpi = w + o; o += NP;
    float* deg_ppi = w + o; o += NP;
    size_t zeroCount = o;
    float* h1a = w + o; o += (size_t)ND * DH;   // x_drug @ W1_ddi
    float* h1b = w + o; o += (size_t)ND * DH;   // x_drug @ W1_dpi
    float* h1c = w + o; o += (size_t)NP * DH;   // x_prot @ W1_ppi
    float* h_d = w + o; o += (size_t)ND * DH;   // layer-1 drug hidden
    float* h_p = w + o; o += (size_t)NP * DH;   // layer-1 protein hidden
    float* h2a = w + o; o += (size_t)ND * DOUT2; // h_d @ W2_ddi
    float* h2b = w + o; o += (size_t)ND * DOUT2; // h_d @ W2_dpi
    float* h2c = w + o; o += (size_t)NP * DOUT2; // h_p @ W2_ppi

    // 1) zero accumulators + degrees
    zero_kernel<<<2048, 256, 0, stream>>>(w, zeroCount);

    // 2) degrees (shared by both layers: same dst index arrays)
    const int eb = (NE + 255) / 256;
    degree_kernel<<<eb, 256, 0, stream>>>(dst_ddi, deg_ddi, NE);
    degree_kernel<<<eb, 256, 0, stream>>>(dst_dpi, deg_dpi, NE);
    degree_kernel<<<eb, 256, 0, stream>>>(dst_ppi, deg_ppi, NE);

    // 3) layer-1 dense transforms (WMMA f32, 16-row strip per wave)
    const int rowGrid = ((ND / 16) + 7) / 8;   // 8 waves per 256-thread block
    gemm_wmma_f32<DIN, DH><<<rowGrid, 256, 0, stream>>>(x_drug, W1_ddi, h1a, ND);
    gemm_wmma_f32<DIN, DH><<<rowGrid, 256, 0, stream>>>(x_drug, W1_dpi, h1b, ND);
    gemm_wmma_f32<DIN, DH><<<rowGrid, 256, 0, stream>>>(x_prot, W1_ppi, h1c, NP);

    // 4) layer-1 edge aggregation (warp per edge, prefetch + native f32 atomics)
    scatter_add<DH><<<1024, 256, 0, stream>>>(h1a, src_ddi, dst_ddi, aggd,  NE);
    scatter_add<DH><<<1024, 256, 0, stream>>>(h1b, src_dpi, dst_dpi, aggp1, NE);
    scatter_add<DH><<<1024, 256, 0, stream>>>(h1c, src_ppi, dst_ppi, aggp2, NE);

    // 5) normalize + bias (protein side fuses the two-relation sum)
    {
        int g1 = (int)(((size_t)ND * DH + 255) / 256);
        int g2 = (int)(((size_t)NP * DH + 255) / 256);
        finalize1<DH><<<g1, 256, 0, stream>>>(aggd, deg_ddi, b1_ddi, h_d, ND);
        finalize2<DH><<<g2, 256, 0, stream>>>(aggp1, deg_dpi, b1_dpi,
                                              aggp2, deg_ppi, b1_ppi, h_p, NP);
    }

    // 6) layer-2 dense transforms (WMMA f32)
    gemm_wmma_f32<DH, DOUT2><<<rowGrid, 256, 0, stream>>>(h_d, W2_ddi, h2a, ND);
    gemm_wmma_f32<DH, DOUT2><<<rowGrid, 256, 0, stream>>>(h_d, W2_dpi, h2b, ND);
    gemm_wmma_f32<DH, DOUT2><<<rowGrid, 256, 0, stream>>>(h_p, W2_ppi, h2c, NP);

    // 7) layer-2 edge aggregation
    scatter_add<DOUT2><<<1024, 256, 0, stream>>>(h2a, src_ddi, dst_ddi, agg2d,  NE);
    scatter_add<DOUT2><<<1024, 256, 0, stream>>>(h2b, src_dpi, dst_dpi, agg2p1, NE);
    scatter_add<DOUT2><<<1024, 256, 0, stream>>>(h2c, src_ppi, dst_ppi, agg2p2, NE);

    // 8) final outputs: [o_d ; o_p] concatenated
    {
        int g1 = (int)(((size_t)ND * DOUT2 + 255) / 256);
        int g2 = (int)(((size_t)NP * DOUT2 + 255) / 256);
        finalize1<DOUT2><<<g1, 256, 0, stream>>>(agg2d, deg_ddi, b2_ddi, out, ND);
        finalize2<DOUT2><<<g2, 256, 0, stream>>>(agg2p1, deg_dpi, b2_dpi,
                                                 agg2p2, deg_ppi, b2_ppi,
                                                 out + (size_t)ND * DOUT2, NP);
    }
}